// MatrixAttention_21869973471319
// MI455X (gfx1250) — compile-verified
//
#include <hip/hip_runtime.h>
#include <math.h>

typedef __attribute__((ext_vector_type(2))) float v2f;
typedef __attribute__((ext_vector_type(8))) float v8f;

#define HW 4096   // 64*64 pixels per image plane
#define WD 64

static __device__ __forceinline__ v8f wmma4(v2f a, v2f b, v8f c) {
  // V_WMMA_F32_16X16X4_F32 : D = A(16x4) * B(4x16) + C(16x16), full fp32
  return __builtin_amdgcn_wmma_f32_16x16x4_f32(false, a, false, b, (short)0, c,
                                               false, false);
}

static __device__ __forceinline__ v8f zero8() {
  v8f z = {0.f, 0.f, 0.f, 0.f, 0.f, 0.f, 0.f, 0.f};
  return z;
}

// ---------------------------------------------------------------------------
// GEMM + fused BN:  out[b, m, p] = (sum_k A[m,k] * B[b, k, p]) * scale[m] + bias[m]
// A row-major [M,K]; B [NB, K, 4096]; out [NB, M, 4096].
// One wave computes a 16x64 strip (A fragment reused over 4 N-tiles).
// ---------------------------------------------------------------------------
__global__ __launch_bounds__(256) void gemm_bn_kernel(
    const float* __restrict__ A, const float* __restrict__ Bm,
    const float* __restrict__ scale, const float* __restrict__ bias,
    float* __restrict__ out, int M, int K) {
  const int wave = (blockIdx.x * blockDim.x + threadIdx.x) >> 5;
  const int lane = threadIdx.x & 31;
  const int m16 = lane & 15;
  const int half = lane >> 4;
  const int Mt = M >> 4;

  int t = wave;
  const int ns = t & 63; t >>= 6;          // 64 strips of 64 columns
  const int mt = t % Mt; t /= Mt;
  const int b  = t;

  const int row = mt * 16 + m16;
  const float* Ap = A + (size_t)row * K;
  const float* Bp = Bm + (size_t)b * K * HW + ns * 64 + m16;

  v8f acc0 = zero8(), acc1 = zero8(), acc2 = zero8(), acc3 = zero8();

  for (int k0 = 0; k0 < K; k0 += 4) {
    const int ka = k0 + 2 * half;
    v2f a;
    a.x = Ap[ka];
    a.y = Ap[ka + 1];
    const float* bp = Bp + (size_t)ka * HW;
    v2f b0, b1, b2, b3;
    b0.x = bp[0];  b0.y = bp[HW + 0];
    b1.x = bp[16]; b1.y = bp[HW + 16];
    b2.x = bp[32]; b2.y = bp[HW + 32];
    b3.x = bp[48]; b3.y = bp[HW + 48];
    acc0 = wmma4(a, b0, acc0);
    acc1 = wmma4(a, b1, acc1);
    acc2 = wmma4(a, b2, acc2);
    acc3 = wmma4(a, b3, acc3);
  }

  const size_t obase = (size_t)b * M * HW;
  const int col = ns * 64 + m16;
#pragma unroll
  for (int r = 0; r < 8; ++r) {
    const int m = mt * 16 + r + 8 * half;
    const float s = scale[m];
    const float bi = bias[m];
    float* o = out + obase + (size_t)m * HW + col;
    o[0]  = acc0[r] * s + bi;
    o[16] = acc1[r] * s + bi;
    o[32] = acc2[r] * s + bi;
    o[48] = acc3[r] * s + bi;
  }
}

// ---------------------------------------------------------------------------
// Attention logits (64x64, K=16) + softmax over M axis. Parametrized for
// row-attention (per (b,n,w)) and col-attention (per (b,n,h)).
//   A(m,k) = a_base[k*4096 + m*mstride], B(k,n) = b_base[k*4096 + n*mstride]
//   out at o_base[m*o_m + col*o_n]
// One wave owns the full 64x64 tile: acc[4][4] (128 VGPRs).
// ---------------------------------------------------------------------------
__global__ __launch_bounds__(128) void attn_softmax_kernel(
    const float* __restrict__ rcv, float* __restrict__ obuf,
    int a_ch_off, int b_ch_off, int mstride, int fix_mul,
    int o_m, int o_n, int o_fix) {
  const int wave = (blockIdx.x * blockDim.x + threadIdx.x) >> 5;
  const int lane = threadIdx.x & 31;
  const int m16 = lane & 15;
  const int half = lane >> 4;

  const int q = wave & 63;   // w (row) or h (col)
  const int bn = wave >> 6;  // 0..15 : b*8 + n
  const int bq = bn >> 3;
  const int nh = bn & 7;
  const int ch_base = bq * 768 + nh * 96;

  const float* a_base = rcv + (size_t)(ch_base + a_ch_off) * HW + q * fix_mul;
  const float* b_base = rcv + (size_t)(ch_base + b_ch_off) * HW + q * fix_mul;
  float* o_base = obuf + (size_t)bn * 64 * HW + (size_t)q * o_fix;

  v8f acc[4][4];
#pragma unroll
  for (int i = 0; i < 4; ++i)
#pragma unroll
    for (int j = 0; j < 4; ++j) acc[i][j] = zero8();

#pragma unroll
  for (int k0 = 0; k0 < 16; k0 += 4) {
    const int ka = k0 + 2 * half;
    v2f af[4], bf[4];
#pragma unroll
    for (int tt = 0; tt < 4; ++tt) {
      const int mm = (tt * 16 + m16) * mstride;
      af[tt].x = a_base[(size_t)ka * HW + mm];
      af[tt].y = a_base[(size_t)(ka + 1) * HW + mm];
      bf[tt].x = b_base[(size_t)ka * HW + mm];
      bf[tt].y = b_base[(size_t)(ka + 1) * HW + mm];
    }
#pragma unroll
    for (int mt = 0; mt < 4; ++mt)
#pragma unroll
      for (int nt = 0; nt < 4; ++nt)
        acc[mt][nt] = wmma4(af[mt], bf[nt], acc[mt][nt]);
  }

  // softmax over the M (i) axis for each column; SCALE = 16^-0.5 = 0.25
  const float SC = 0.25f;
#pragma unroll
  for (int nt = 0; nt < 4; ++nt) {
    float mx = -INFINITY;
#pragma unroll
    for (int mt = 0; mt < 4; ++mt)
#pragma unroll
      for (int r = 0; r < 8; ++r) mx = fmaxf(mx, acc[mt][nt][r]);
    mx = fmaxf(mx, __shfl_xor(mx, 16, 32));
    float sum = 0.f;
#pragma unroll
    for (int mt = 0; mt < 4; ++mt)
#pragma unroll
      for (int r = 0; r < 8; ++r) {
        const float e = __expf(SC * (acc[mt][nt][r] - mx));
        acc[mt][nt][r] = e;
        sum += e;
      }
    sum += __shfl_xor(sum, 16, 32);
    const float inv = 1.0f / sum;
#pragma unroll
    for (int mt = 0; mt < 4; ++mt)
#pragma unroll
      for (int r = 0; r < 8; ++r) acc[mt][nt][r] *= inv;
  }

#pragma unroll
  for (int mt = 0; mt < 4; ++mt)
#pragma unroll
    for (int nt = 0; nt < 4; ++nt) {
      const int col = nt * 16 + m16;
#pragma unroll
      for (int r = 0; r < 8; ++r) {
        const int m = mt * 16 + r + 8 * half;
        o_base[(size_t)m * o_m + (size_t)col * o_n] = acc[mt][nt][r];
      }
    }
}

// ---------------------------------------------------------------------------
// Combine: one wave per (b,n,h,d).
//   G[i,w] = sum_j v[d,i,j] * attn_c[j,h,w]   (64x64 GEMM, K=64, 256 WMMA)
//   out[d,h,w] = sum_i attn_r[i,h,w] * G[i,w] (VALU + half-shuffle reduce)
// ---------------------------------------------------------------------------
__global__ __launch_bounds__(256) void combine_kernel(
    const float* __restrict__ rcv, const float* __restrict__ ar,
    const float* __restrict__ ac, float* __restrict__ outp) {
  const int wave = (blockIdx.x * blockDim.x + threadIdx.x) >> 5;
  const int lane = threadIdx.x & 31;
  const int m16 = lane & 15;
  const int half = lane >> 4;

  const int d = wave & 31;
  const int h = (wave >> 5) & 63;
  const int bn = wave >> 11;  // 0..15
  const int bq = bn >> 3;
  const int nh = bn & 7;

  const float* a_base = rcv + (size_t)(bq * 768 + nh * 96 + 64 + d) * HW; // v[d]: [i*64+j]
  const float* b_base = ac + ((size_t)bn * 64 + h) * HW;                  // [j*64+w]
  const float* ar_base = ar + (size_t)bn * 64 * HW + h * 64;              // [i*4096+w]

  v8f acc[4][4];
#pragma unroll
  for (int i = 0; i < 4; ++i)
#pragma unroll
    for (int j = 0; j < 4; ++j) acc[i][j] = zero8();

  for (int k0 = 0; k0 < 64; k0 += 4) {
    const int ka = k0 + 2 * half;
    v2f af[4], bf[4];
#pragma unroll
    for (int tt = 0; tt < 4; ++tt) {
      const float* ap = a_base + (tt * 16 + m16) * 64 + ka;  // contiguous K pair
      af[tt].x = ap[0];
      af[tt].y = ap[1];
      const int col = tt * 16 + m16;
      bf[tt].x = b_base[ka * 64 + col];
      bf[tt].y = b_base[(ka + 1) * 64 + col];
    }
#pragma unroll
    for (int mt = 0; mt < 4; ++mt)
#pragma unroll
      for (int nt = 0; nt < 4; ++nt)
        acc[mt][nt] = wmma4(af[mt], bf[nt], acc[mt][nt]);
  }

  float res[4];
#pragma unroll
  for (int nt = 0; nt < 4; ++nt) {
    const int w = nt * 16 + m16;
    float p = 0.f;
#pragma unroll
    for (int mt = 0; mt < 4; ++mt)
#pragma unroll
      for (int r = 0; r < 8; ++r) {
        const int i = mt * 16 + r + 8 * half;
        p += acc[mt][nt][r] * ar_base[(size_t)i * HW + w];
      }
    p += __shfl_xor(p, 16, 32);
    res[nt] = p;
  }

  if (half == 0) {
    float* o = outp + (size_t)(bq * 256 + nh * 32 + d) * HW + h * 64;
#pragma unroll
    for (int nt = 0; nt < 4; ++nt) o[nt * 16 + m16] = res[nt];
  }
}

// ---------------------------------------------------------------------------
// Positional-encoding branch: out += dwconv3x3(v) * pe_scale + pe_bias
// ---------------------------------------------------------------------------
__global__ __launch_bounds__(256) void pe_kernel(
    const float* __restrict__ rcv, const float* __restrict__ pw,
    const float* __restrict__ ps, const float* __restrict__ pb,
    float* __restrict__ outp) {
  const int idx = blockIdx.x * blockDim.x + threadIdx.x;  // 2*256*4096
  const int w = idx & 63;
  const int h = (idx >> 6) & 63;
  const int c = (idx >> 12) & 255;
  const int b = idx >> 20;
  const int nh = c >> 5, d = c & 31;
  const float* v = rcv + (size_t)(b * 768 + nh * 96 + 64 + d) * HW;
  const float* wt = pw + c * 9;
  float s = 0.f;
#pragma unroll
  for (int kh = 0; kh < 3; ++kh) {
    const int hh = h + kh - 1;
    if (hh < 0 || hh > 63) continue;
#pragma unroll
    for (int kw = 0; kw < 3; ++kw) {
      const int ww = w + kw - 1;
      if (ww < 0 || ww > 63) continue;
      s += v[hh * 64 + ww] * wt[kh * 3 + kw];
    }
  }
  const size_t off = (size_t)(b * 256 + c) * HW + h * 64 + w;
  outp[off] = outp[off] + s * ps[c] + pb[c];
}

// ---------------------------------------------------------------------------
extern "C" void kernel_launch(void* const* d_in, const int* in_sizes, int n_in,
                              void* d_out, int out_size, void* d_ws,
                              size_t ws_size, hipStream_t stream) {
  (void)in_sizes; (void)n_in; (void)out_size; (void)ws_size;
  const float* x         = (const float*)d_in[0];
  const float* rcv_w     = (const float*)d_in[1];
  const float* rcv_scale = (const float*)d_in[2];
  const float* rcv_bias  = (const float*)d_in[3];
  const float* pe_w      = (const float*)d_in[4];
  const float* pe_scale  = (const float*)d_in[5];
  const float* pe_bias   = (const float*)d_in[6];
  const float* proj_w    = (const float*)d_in[7];
  const float* proj_scale= (const float*)d_in[8];
  const float* proj_bias = (const float*)d_in[9];
  float* out = (float*)d_out;

  float* rcv = (float*)d_ws;           // [2, 768, 4096]  = 6,291,456 f32
  float* ar  = rcv + 6291456;          // [16, 64, 64, 64] attn_r (i,h,w)
  float* ac  = ar + 4194304;           // [16, 64, 64, 64] attn_c (h,j,w)
  float* att = ac + 4194304;           // [2, 256, 4096]  pre-proj activations

  // 1) rcv = BN(conv1x1(x)) : M=768, K=256 ; 6144 waves -> 768 blocks x 8 waves
  gemm_bn_kernel<<<dim3(768), dim3(256), 0, stream>>>(
      rcv_w, x, rcv_scale, rcv_bias, rcv, 768, 256);

  // 2) row attention (per (b,n,w)): A stride 64 (i*W), out [i*4096 + h*64 + w]
  attn_softmax_kernel<<<dim3(256), dim3(128), 0, stream>>>(
      rcv, ar, /*a_ch*/0, /*b_ch*/16, /*mstride*/64, /*fix_mul*/1,
      /*o_m*/4096, /*o_n*/64, /*o_fix*/1);

  // 3) col attention (per (b,n,h)): A stride 1 (i contiguous), out [h*4096 + i*64 + j]
  attn_softmax_kernel<<<dim3(256), dim3(128), 0, stream>>>(
      rcv, ac, /*a_ch*/32, /*b_ch*/48, /*mstride*/1, /*fix_mul*/64,
      /*o_m*/64, /*o_n*/1, /*o_fix*/4096);

  // 4) 3-way combine: 32768 waves -> 4096 blocks x 8 waves
  combine_kernel<<<dim3(4096), dim3(256), 0, stream>>>(rcv, ar, ac, att);

  // 5) positional-encoding depthwise 3x3 add
  pe_kernel<<<dim3(8192), dim3(256), 0, stream>>>(rcv, pe_w, pe_scale, pe_bias, att);

  // 6) proj = BN(conv1x1(att)) : M=256, K=256 -> d_out ; 2048 waves
  gemm_bn_kernel<<<dim3(256), dim3(256), 0, stream>>>(
      proj_w, att, proj_scale, proj_bias, out, 256, 256);
}